// Cluster_wise_linear_747324309799
// MI455X (gfx1250) — compile-verified
//
#include <hip/hip_runtime.h>

typedef float v2f __attribute__((ext_vector_type(2)));
typedef float v8f __attribute__((ext_vector_type(8)));
typedef int b128_t __attribute__((vector_size(16)));   // matches builtin's pointee type

#define BSZ 128
#define N_VARS 321
#define IN_DIM 336
#define OUT_DIM 96
#define N_CLUSTER 8
#define M_TOTAL (BSZ * N_VARS)        // 41088 rows

#define KC 48                         // K-chunk: 336 = 7 * 48 (exact)
#define NCHUNKS (IN_DIM / KC)         // 7
#define NQUAD (KC / 4)                // 12 k-quads per chunk
#define LDS_STRIDE 52                 // padded row stride (floats): 208B rows, 16B aligned
#define THREADS 128                   // 4 waves
#define ROWS_PER_BLOCK 128            // 41088 / 128 = 321 blocks, no tail
#define NT_TILES (OUT_DIM / 16)       // 6
#define STAGE_PER_THREAD ((OUT_DIM * NQUAD) / THREADS)   // 9 x b128 per thread per chunk
#define NITER (NCHUNKS * N_CLUSTER)   // 56 (chunk outer, cluster inner)

#if __has_builtin(__builtin_amdgcn_global_load_async_to_lds_b128)
#define HAVE_ASYNC_B128 1
#else
#define HAVE_ASYNC_B128 0
#endif
#if __has_builtin(__builtin_amdgcn_s_wait_asynccnt)
#define HAVE_WAIT_ASYNC 1
#else
#define HAVE_WAIT_ASYNC 0
#endif

// One 16B memory -> LDS async copy (ASYNCcnt-tracked, no VGPR round trip)
__device__ __forceinline__ void async_copy16(const float* g, float* l) {
#if HAVE_ASYNC_B128
    __builtin_amdgcn_global_load_async_to_lds_b128(
        (__attribute__((address_space(1))) b128_t*)g,
        (__attribute__((address_space(3))) b128_t*)l, 0, 0);
#else
    unsigned loff = (unsigned)(unsigned long long)(__attribute__((address_space(3))) void*)l;
    asm volatile("global_load_async_to_lds_b128 %0, %1, off"
                 :: "v"(loff), "v"(g) : "memory");
#endif
}

template <int N>
__device__ __forceinline__ void wait_asynccnt() {
#if HAVE_WAIT_ASYNC
    __builtin_amdgcn_s_wait_asynccnt(N);
#else
    asm volatile("s_wait_asynccnt %0" :: "i"(N) : "memory");
#endif
}

// Issue this thread's 9 async b128 copies of W[c, 0:96, i0:i0+48] -> buf[o][kk]
__device__ __forceinline__ void stage_issue(const float* Wc, int i0, float* buf, int tid) {
#pragma unroll
    for (int rep = 0; rep < STAGE_PER_THREAD; ++rep) {
        const int j  = rep * THREADS + tid;
        const int o  = j / NQUAD;
        const int q4 = j % NQUAD;
        async_copy16(Wc + (size_t)o * IN_DIM + i0 + 4 * q4,
                     buf + o * LDS_STRIDE + 4 * q4);
    }
}

__global__ __launch_bounds__(THREADS)
void cluster_wise_linear_wmma(const float* __restrict__ x,     // (M, 336)
                              const float* __restrict__ prob,  // (M, 8)
                              const float* __restrict__ W,     // (8, 96, 336)
                              const float* __restrict__ bias,  // (8, 96)
                              float* __restrict__ out)         // (M, 96)
{
    __shared__ float ldsB[2][OUT_DIM * LDS_STRIDE];   // double buffer, 2 x 19968 B

    const int tid  = threadIdx.x;
    const int wave = tid >> 5;
    const int lane = tid & 31;
    const int l16  = lane & 15;
    const int hi   = lane >> 4;                        // 0: K pair {0,1}; 1: K pair {2,3}
    const int m0   = blockIdx.x * ROWS_PER_BLOCK + wave * 32;

    v8f acc[2][6];
    const v8f vzero = {0.f, 0.f, 0.f, 0.f, 0.f, 0.f, 0.f, 0.f};
#pragma unroll
    for (int mt = 0; mt < 2; ++mt)
#pragma unroll
        for (int nt = 0; nt < NT_TILES; ++nt) acc[mt][nt] = vzero;

    const float* xrow0 = x + (size_t)(m0 + l16) * IN_DIM;
    const float* xrow1 = x + (size_t)(m0 + 16 + l16) * IN_DIM;
    const float* prow0 = prob + (size_t)(m0 + l16) * N_CLUSTER;
    const float* prow1 = prob + (size_t)(m0 + 16 + l16) * N_CLUSTER;

    // x fragments for the current chunk, held in registers across all 8 clusters
    v2f xa0[NQUAD], xa1[NQUAD];

    // Prologue: kick off staging of (chunk 0, cluster 0)
    stage_issue(W, 0, ldsB[0], tid);

    for (int t = 0; t < NITER; ++t) {
        const int chunk = t >> 3;          // outer: K chunk
        const int c     = t & 7;           // inner: cluster
        const int i0    = chunk * KC;

        // Issue next tile's async copies into the idle buffer
        if (t + 1 < NITER) {
            const int c1  = (t + 1) & 7;
            const int ch1 = (t + 1) >> 3;
            stage_issue(W + (size_t)c1 * OUT_DIM * IN_DIM, ch1 * KC, ldsB[(t + 1) & 1], tid);
        }

        // Refresh x fragments once per chunk (x streamed exactly once overall)
        if (c == 0) {
#pragma unroll
            for (int q = 0; q < NQUAD; ++q) {
                const int kb = 4 * q + 2 * hi;
                xa0[q] = *(const v2f*)(xrow0 + i0 + kb);
                xa1[q] = *(const v2f*)(xrow1 + i0 + kb);
            }
            if (chunk + 1 < NCHUNKS) {
                __builtin_prefetch(xrow0 + i0 + KC, 0, 1);
                __builtin_prefetch(xrow1 + i0 + KC, 0, 1);
            }
        }

        const float p0 = prow0[c];
        const float p1 = prow1[c];

        // In-order async completion: <=9 outstanding means current buffer is ready
        if (t + 1 < NITER) wait_asynccnt<STAGE_PER_THREAD>();
        else               wait_asynccnt<0>();
        __syncthreads();

        const float* bufc = ldsB[t & 1];
#pragma unroll
        for (int q = 0; q < NQUAD; ++q) {
            const int kb = 4 * q + 2 * hi;
            v2f a0 = xa0[q] * p0;
            v2f a1 = xa1[q] * p1;
#pragma unroll
            for (int nt = 0; nt < NT_TILES; ++nt) {
                const v2f bf = *(const v2f*)(&bufc[(nt * 16 + l16) * LDS_STRIDE + kb]);
                acc[0][nt] = __builtin_amdgcn_wmma_f32_16x16x4_f32(
                    false, a0, false, bf, (short)0, acc[0][nt], false, false);
                acc[1][nt] = __builtin_amdgcn_wmma_f32_16x16x4_f32(
                    false, a1, false, bf, (short)0, acc[1][nt], false, false);
            }
        }
        __syncthreads();   // all waves done reading buf[t&1] before it is re-staged at t+1
    }

    // Bias folded in as one rank-1 k-quad per cluster: A col = prob[m,c], B row = b[c,:]
    for (int c = 0; c < N_CLUSTER; ++c) {
        const float p0 = prow0[c];
        const float p1 = prow1[c];
        const v2f a0b = {hi ? 0.f : p0, 0.f};
        const v2f a1b = {hi ? 0.f : p1, 0.f};
#pragma unroll
        for (int nt = 0; nt < NT_TILES; ++nt) {
            const float bv = bias[(size_t)c * OUT_DIM + nt * 16 + l16];
            const v2f bf = {hi ? 0.f : bv, 0.f};
            acc[0][nt] = __builtin_amdgcn_wmma_f32_16x16x4_f32(
                false, a0b, false, bf, (short)0, acc[0][nt], false, false);
            acc[1][nt] = __builtin_amdgcn_wmma_f32_16x16x4_f32(
                false, a1b, false, bf, (short)0, acc[1][nt], false, false);
        }
    }

    // C layout: VGPR r, lanes 0-15 -> row r, lanes 16-31 -> row r+8; col = l16
#pragma unroll
    for (int mt = 0; mt < 2; ++mt) {
        const int rbase = m0 + mt * 16 + 8 * hi;
#pragma unroll
        for (int nt = 0; nt < NT_TILES; ++nt) {
#pragma unroll
            for (int r = 0; r < 8; ++r) {
                out[(size_t)(rbase + r) * OUT_DIM + nt * 16 + l16] = acc[mt][nt][r];
            }
        }
    }
}

extern "C" void kernel_launch(void* const* d_in, const int* in_sizes, int n_in,
                              void* d_out, int out_size, void* d_ws, size_t ws_size,
                              hipStream_t stream) {
    (void)in_sizes; (void)n_in; (void)out_size; (void)d_ws; (void)ws_size;
    const float* x    = (const float*)d_in[0];
    const float* prob = (const float*)d_in[1];
    const float* W    = (const float*)d_in[2];
    const float* b    = (const float*)d_in[3];
    float* out        = (float*)d_out;

    dim3 grid(M_TOTAL / ROWS_PER_BLOCK);   // 321
    dim3 block(THREADS);                   // 128 = 4 waves
    hipLaunchKernelGGL(cluster_wise_linear_wmma, grid, block, 0, stream,
                       x, prob, W, b, out);
}